// DetectionLoss_34256659153485
// MI455X (gfx1250) — compile-verified
//
#include <hip/hip_runtime.h>
#include <cstdint>

// RetinaNet-style detection loss for MI455X (gfx1250, wave32).
// VALU-bound (IoU loop ~8e8 lane-ops) over a ~44MB / ~1.9us memory floor, so
// this version removes the per-pair f32 divide (division-free argmax) and
// hoists per-GT areas into LDS.
// CDNA5 paths: async global->LDS staging of the GT tile, s_wait_asynccnt,
// wave32 shuffle reductions, global_atomic_add_f64 accumulation.

#define GTN 32        // G: gt boxes per image (fixed by reference)
#define NCLS 8        // NUM_CLASSES
static constexpr float POS_IOU = 0.5f;
static constexpr float ALPHA   = 0.25f;
static constexpr float BETA_SL = 1.0f / 9.0f;
static constexpr float EPSF    = 1e-7f;

struct Accum {
    double focal;     // sum of focal loss terms
    double sl1;       // sum of smooth-L1 terms (positives only)
    unsigned npos;    // number of positive anchors
    unsigned pad;
};

__global__ void detloss_init_ws(Accum* ws) {
    ws->focal = 0.0;
    ws->sl1   = 0.0;
    ws->npos  = 0u;
    ws->pad   = 0u;
}

__global__ __launch_bounds__(256) void detloss_main(
    const float* __restrict__ cls_preds,   // [B, A, 8]
    const float* __restrict__ reg_preds,   // [B, A, 4]
    const float* __restrict__ anchors,     // [A, 4]
    const float* __restrict__ gt_boxes,    // [B, 32, 4]
    const int*   __restrict__ gt_labels,   // [B, 32]
    Accum* __restrict__ ws,
    int A)
{
    __shared__ alignas(16) float s_gt[GTN * 4];
    __shared__ float s_area[GTN];
    __shared__ int   s_lbl[GTN];
    __shared__ float s_red_f[8];
    __shared__ float s_red_s[8];
    __shared__ unsigned s_red_n[8];

    const int b    = blockIdx.y;
    const int a    = blockIdx.x * blockDim.x + threadIdx.x;
    const int lane = threadIdx.x & 31;
    const int wid  = threadIdx.x >> 5;

    // ---- Stage this image's GT boxes (512B) + labels (128B) into LDS via the
    // CDNA5 async global->LDS path. Every wave issues the same 32x(b128)+32x(b32)
    // loads redundantly so its own s_wait_asynccnt 0 guarantees visibility of
    // its copies; duplicate writes carry identical bytes.
    {
        uint32_t lds_box = (uint32_t)(uintptr_t)(&s_gt[lane * 4]);
        uint32_t lds_lbl = (uint32_t)(uintptr_t)(&s_lbl[lane]);
        uint32_t off_box = (uint32_t)((b * GTN + lane) * 16); // 4 floats/box
        uint32_t off_lbl = (uint32_t)((b * GTN + lane) * 4);
        asm volatile(
            "global_load_async_to_lds_b128 %0, %1, %2\n\t"
            "global_load_async_to_lds_b32  %3, %4, %5\n\t"
            "s_wait_asynccnt 0"
            :
            : "v"(lds_box), "v"(off_box), "s"(gt_boxes),
              "v"(lds_lbl), "v"(off_lbl), "s"(gt_labels)
            : "memory");
    }
    __syncthreads();

    // Precompute per-GT areas once per block (saves 3 VALU ops per IoU pair).
    if (threadIdx.x < GTN) {
        const float4 gb = ((const float4*)s_gt)[threadIdx.x];
        s_area[threadIdx.x] = (gb.z - gb.x) * (gb.w - gb.y);
    }
    __syncthreads();

    float focal_sum = 0.0f;
    float sl1_sum   = 0.0f;
    unsigned pos_c  = 0u;

    if (a < A) {
        // Anchor box (coalesced b128 load)
        const float4 an = ((const float4*)anchors)[a];
        const float wa = an.z - an.x;
        const float ha = an.w - an.y;
        const float xa = (an.x + an.z) * 0.5f;
        const float ya = (an.y + an.w) * 0.5f;
        const float area_a = wa * ha;

        // Division-free first-occurrence argmax over 32 GT boxes:
        // iou_g > iou_best  <=>  inter_g*denom_best > inter_best*denom_g
        // (denoms > 0). Init (-1, 1) reproduces best_iou = -1 semantics.
        float best_inter = -1.0f;
        float best_denom = 1.0f;
        int   best       = 0;
        #pragma unroll 8
        for (int g = 0; g < GTN; ++g) {
            const float4 gb = ((const float4*)s_gt)[g];   // LDS broadcast b128
            const float lx = fmaxf(an.x, gb.x);
            const float ly = fmaxf(an.y, gb.y);
            const float rx = fminf(an.z, gb.z);
            const float ry = fminf(an.w, gb.w);
            const float iw = fmaxf(rx - lx, 0.0f);
            const float ih = fmaxf(ry - ly, 0.0f);
            const float inter = iw * ih;
            const float denom = (area_a + s_area[g]) - inter;
            const bool better = inter * best_denom > best_inter * denom;
            best_inter = better ? inter : best_inter;
            best_denom = better ? denom : best_denom;
            best       = better ? g     : best;
        }
        // One divide total: exactly the reference's rounding for the best box.
        const float max_iou = best_inter / best_denom;
        const bool pos   = max_iou >= POS_IOU;
        const int  label = s_lbl[best];

        // Focal loss over 8 classes (two b128 loads)
        const float4* cp = (const float4*)(cls_preds + ((size_t)b * A + a) * NCLS);
        const float4 c0 = cp[0];
        const float4 c1 = cp[1];
        float x8[NCLS] = { c0.x, c0.y, c0.z, c0.w, c1.x, c1.y, c1.z, c1.w };
        #pragma unroll
        for (int c = 0; c < NCLS; ++c) {
            const float t  = (pos && (c == label)) ? 1.0f : 0.0f;
            const float x  = x8[c];
            const float ey = __expf(-fabsf(x));
            const float bce = fmaxf(x, 0.0f) - x * t + __logf(1.0f + ey);
            const float pt  = __expf(-bce);
            const float at  = (t == 1.0f) ? ALPHA : (1.0f - ALPHA);
            const float om  = 1.0f - pt;
            focal_sum += at * om * om * bce;
        }

        // Smooth-L1 over positive anchors only (reg_preds read only when pos:
        // keeps ~20MB of reg traffic off the bus for typical npos fractions)
        if (pos) {
            pos_c = 1u;
            const float4 gb = ((const float4*)s_gt)[best];
            const float wg = gb.z - gb.x;
            const float hg = gb.w - gb.y;
            const float xg = (gb.x + gb.z) * 0.5f;
            const float yg = (gb.y + gb.w) * 0.5f;
            const float tx = (xg - xa) / (wa + EPSF);
            const float ty = (yg - ya) / (ha + EPSF);
            const float tw = __logf(wg / (wa + EPSF) + EPSF);
            const float th = __logf(hg / (ha + EPSF) + EPSF);
            const float4 rp = ((const float4*)reg_preds)[(size_t)b * A + a];
            const float d0 = fabsf(rp.x - tx);
            const float d1 = fabsf(rp.y - ty);
            const float d2 = fabsf(rp.z - tw);
            const float d3 = fabsf(rp.w - th);
            const float s0 = (d0 < BETA_SL) ? 0.5f * d0 * d0 / BETA_SL : d0 - 0.5f * BETA_SL;
            const float s1 = (d1 < BETA_SL) ? 0.5f * d1 * d1 / BETA_SL : d1 - 0.5f * BETA_SL;
            const float s2 = (d2 < BETA_SL) ? 0.5f * d2 * d2 / BETA_SL : d2 - 0.5f * BETA_SL;
            const float s3 = (d3 < BETA_SL) ? 0.5f * d3 * d3 / BETA_SL : d3 - 0.5f * BETA_SL;
            sl1_sum = s0 + s1 + s2 + s3;
        }
    }

    // wave32 tree reduction
    #pragma unroll
    for (int o = 16; o > 0; o >>= 1) {
        focal_sum += __shfl_down(focal_sum, o, 32);
        sl1_sum   += __shfl_down(sl1_sum,   o, 32);
        pos_c     += __shfl_down(pos_c,     o, 32);
    }
    if (lane == 0) {
        s_red_f[wid] = focal_sum;
        s_red_s[wid] = sl1_sum;
        s_red_n[wid] = pos_c;
    }
    __syncthreads();
    if (threadIdx.x == 0) {
        double f = 0.0, s = 0.0;
        unsigned n = 0u;
        const int nw = (int)(blockDim.x + 31u) >> 5;
        for (int i = 0; i < nw; ++i) { f += (double)s_red_f[i]; s += (double)s_red_s[i]; n += s_red_n[i]; }
        atomicAdd(&ws->focal, f);     // global_atomic_add_f64
        atomicAdd(&ws->sl1,   s);
        atomicAdd(&ws->npos,  n);
    }
}

__global__ void detloss_finalize(const Accum* __restrict__ ws, float* __restrict__ out) {
    const float npos  = (float)ws->npos;
    const float denom = fmaxf(npos, 1.0f);
    out[0] = (npos > 0.0f) ? (float)(ws->focal / (double)denom) : 0.0f;
    out[1] = (npos > 0.0f) ? (float)(ws->sl1   / (double)denom) : 0.0f;
}

extern "C" void kernel_launch(void* const* d_in, const int* in_sizes, int n_in,
                              void* d_out, int out_size, void* d_ws, size_t ws_size,
                              hipStream_t stream) {
    const float* cls_preds = (const float*)d_in[0];
    const float* reg_preds = (const float*)d_in[1];
    const float* anchors   = (const float*)d_in[2];
    const float* gt_boxes  = (const float*)d_in[3];
    const int*   gt_labels = (const int*)d_in[4];

    const int A = in_sizes[2] / 4;     // anchors: [A,4]
    const int B = in_sizes[4] / GTN;   // gt_labels: [B,32]

    Accum* ws = (Accum*)d_ws;
    detloss_init_ws<<<1, 1, 0, stream>>>(ws);

    dim3 block(256);
    dim3 grid((A + 255) / 256, B);
    detloss_main<<<grid, block, 0, stream>>>(cls_preds, reg_preds, anchors,
                                             gt_boxes, gt_labels, ws, A);

    detloss_finalize<<<1, 1, 0, stream>>>(ws, (float*)d_out);
}